// HomographEdgeEncoder_72327249264836
// MI455X (gfx1250) — compile-verified
//
#include <hip/hip_runtime.h>

// ---------------------------------------------------------------------------
// HomographEdgeEncoder for MI455X (gfx1250, wave32).
//
// proj (+bias) computed as a GEMM via chained V_WMMA_F32_16X16X4_F32 using a
// block-diagonal "expanded K" weight matrix shared across mixed-type edges:
//   cols 0..59  = concat of per-type continuous slots (d_COL_OFF)
//   cols 60..68 = per-type bias one-hot columns (X[row][60+t] = 1.0)
//   K padded to 72 = 18 chunks of 4; LDS row stride 76 dwords (conflict-free).
// Discrete-embedding gathers (global_load_b32 via AS(1) pointers) issue before
// the WMMA chain and are folded in after it; tile body is templated on FULL so
// full tiles carry zero guards; output streamed with NT stores from a single
// base address + immediate offsets (512 MB output >> 192 MB L2).
// ---------------------------------------------------------------------------

typedef __attribute__((ext_vector_type(2))) float v2f;  // A/B fragment: 16x4 f32 = 2 VGPR
typedef __attribute__((ext_vector_type(8))) float v8f;  // C/D fragment: 16x16 f32 = 8 VGPR
typedef __attribute__((address_space(1))) const float* gcfloat;  // global-AS load ptr

#define KPAD  76           // LDS row stride in dwords; 76 mod 64 = 12 -> conflict-free b64
#define KCH   18           // 18 chunks of K=4 => logical K = 72 (60 cont + 9 bias + pad)
#define KBIAS 60           // first bias one-hot column

// ---- compile-time metadata derived from the reference config ----
__constant__ int d_CONT_N[9]    = {4,7,7,6,7,8,7,8,6};
__constant__ int d_CONT_F[9][8] = {
  {3,6,7,8,0,0,0,0},
  {2,3,4,5,6,7,8,0},
  {2,3,4,5,6,7,8,0},
  {1,4,5,6,7,8,0,0},
  {2,3,4,5,6,7,8,0},
  {1,2,3,4,5,6,7,8},
  {2,3,4,5,6,7,8,0},
  {1,2,3,4,5,6,7,8},
  {0,1,4,6,7,8,0,0}};
__constant__ int d_COL_OFF[9]   = {0,4,11,18,24,31,39,46,54};
__constant__ int d_ND[9]        = {5,2,2,3,2,1,2,1,3};
__constant__ int d_DW[9][5] = {
  {52,51,51,51,51},
  {128,128,0,0,0},
  {128,128,0,0,0},
  {86,85,85,0,0},
  {128,128,0,0,0},
  {256,0,0,0,0},
  {128,128,0,0,0},
  {256,0,0,0,0},
  {86,85,85,0,0}};
__constant__ int d_DF[9][5] = {
  {0,1,2,4,5},
  {0,1,0,0,0},
  {0,1,0,0,0},
  {0,2,3,0,0},
  {0,1,0,0,0},
  {0,0,0,0,0},
  {0,1,0,0,0},
  {0,0,0,0,0},
  {2,3,5,0,0}};
__constant__ int d_TBASE[9]     = {0,5,7,9,12,14,15,17,18};

struct EncParams {
  const float* edge_attr;   // [E,10]
  const float* W[9];        // [256, in_dim_t] row-major
  const float* b[9];        // [256]
  const float* emb[21];     // [rows_t_f, width_t_f]
  float*       out;         // [E,256]
  int          E;
  int          ntiles;      // ceil(E/16)
};

// LDS layout (dword offsets)
#define L_WL   0                         // 256*KPAD : W_big rows [ch][k]
#define L_X    (256*KPAD)                // 16*KPAD  : expanded feature tile
#define L_AI   (L_X + 16*KPAD)           // 16*12    : int-converted attrs
#define L_TY   (L_AI + 16*12)            // 16       : per-row type (-1 invalid)
#define L_META (L_TY + 16)               // 9*256    : packed gather metadata
#define L_EP   (L_META + 9*256)          // 21*2     : emb base pointers (u64)
#define L_TOTAL_DW (L_EP + 48)

template<bool FULL>
__device__ __forceinline__ void tile_body(
    const EncParams& p, int tile, int gridx,
    float* WL, float* X, int* AI, int* TY,
    const unsigned* META, const unsigned long long* EP,
    int tid, int wave, int half, int l16)
{
  // ---- zero X tile (scatter only fills the row's own type slots) ----
  for (int i = tid; i < 16 * KPAD; i += 256) X[i] = 0.0f;
  if (!FULL) { if (tid < 16) TY[tid] = -1; }
  __syncthreads();

  // ---- scatter 16 edges into expanded-K layout; prefetch this block's next tile ----
  if (tid < 16) {
    const int e = tile * 16 + tid;
    if (FULL || e < p.E) {
      const float* a = p.edge_attr + (size_t)e * 10;
      float av[10];
      #pragma unroll
      for (int f = 0; f < 10; ++f) av[f] = a[f];
      const int t = (int)av[9];
      TY[tid] = t;
      #pragma unroll
      for (int f = 0; f < 9; ++f) AI[tid * 12 + f] = (int)av[f];
      const int off = d_COL_OFF[t];
      const int n   = d_CONT_N[t];
      for (int j = 0; j < n; ++j)
        X[tid * KPAD + off + j] = av[d_CONT_F[t][j]];
      X[tid * KPAD + KBIAS + t] = 1.0f;          // bias selector column
    }
    const int en = (tile + gridx) * 16 + tid;
    if (en < p.E) __builtin_prefetch(p.edge_attr + (size_t)en * 10, 0, 1);
  }
  __syncthreads();

  // ---- A fragments: 16x4 f32 layout (row = lane&15, K half by lane>=16) ----
  v2f afrag[KCH];
  const float* xrow = X + l16 * KPAD + 2 * half;
  #pragma unroll
  for (int c = 0; c < KCH; ++c) afrag[c] = *(const v2f*)(xrow + 4 * c);

  int ty[8];
  #pragma unroll
  for (int v = 0; v < 8; ++v) ty[v] = TY[v + 8 * half];

  // one base address per lane; all 16 stores use immediate offsets
  float* base = p.out + ((size_t)(tile * 16 + 8 * half)) * 256 + wave * 32 + l16;

  #pragma unroll
  for (int nti = 0; nti < 2; ++nti) {
    const int ch = wave * 32 + nti * 16 + l16;

    // ---- issue discrete-embedding gathers first (overlap with WMMA chain) ----
    float dval[8];
    #pragma unroll
    for (int v = 0; v < 8; ++v) {
      if (FULL || ty[v] >= 0) {
        const unsigned m = META[ty[v] * 256 + ch];
        const int tbl    = m & 31;
        const int f      = (m >> 5) & 15;
        const int width  = (m >> 9) & 511;
        const int choff  = (int)(m >> 18);
        const int idx    = AI[(v + 8 * half) * 12 + f];
        gcfloat ebase    = (gcfloat)(unsigned long long)EP[tbl];
        dval[v] = ebase[idx * width + choff];    // global_load_b32 (L2-hot tables)
      } else {
        dval[v] = 0.0f;
      }
    }

    // ---- 18 chained V_WMMA_F32_16X16X4_F32 on zero-seeded C: D = X * W_big ----
    v8f acc = {};
    const float* wrow = WL + ch * KPAD + 2 * half;
    #pragma unroll
    for (int c = 0; c < KCH; ++c) {
      v2f bfrag = *(const v2f*)(wrow + 4 * c);
      acc = __builtin_amdgcn_wmma_f32_16x16x4_f32(
          false, afrag[c], false, bfrag, (short)0, acc, false, false);
    }

    // ---- fold gathered embeddings in after the chain; NT streaming stores ----
    #pragma unroll
    for (int v = 0; v < 8; ++v) acc[v] += dval[v];

    if (FULL) {
      #pragma unroll
      for (int v = 0; v < 8; ++v)
        __builtin_nontemporal_store(acc[v], base + (size_t)(v * 256 + nti * 16));
    } else {
      #pragma unroll
      for (int v = 0; v < 8; ++v)
        if (ty[v] >= 0)
          __builtin_nontemporal_store(acc[v], base + (size_t)(v * 256 + nti * 16));
    }
  }
  __syncthreads();   // X/AI/TY reused next iteration
}

__global__ __launch_bounds__(256)
void HomographEdgeEncoder_wmma_kernel(EncParams p) {
  extern __shared__ float smem[];
  float*              WL   = smem + L_WL;
  float*              X    = smem + L_X;
  int*                AI   = (int*)(smem + L_AI);
  int*                TY   = (int*)(smem + L_TY);
  unsigned*           META = (unsigned*)(smem + L_META);
  unsigned long long* EP   = (unsigned long long*)(smem + L_EP);

  const int tid  = threadIdx.x;
  const int lane = tid & 31;
  const int wave = tid >> 5;
  const int half = lane >> 4;          // f32 WMMA: lanes 0-15 carry K0/K1, 16-31 K2/K3
  const int l16  = lane & 15;

  // ---- one-time per-block setup: W_big rows, bias columns, gather metadata ----
  {
    const int ch = tid;                // 256 threads == 256 output channels
    float* row = WL + ch * KPAD;
    #pragma unroll 4
    for (int k = 0; k < KPAD; ++k) row[k] = 0.0f;
    for (int t = 0; t < 9; ++t) {
      const int n = d_CONT_N[t];
      const float* Wt = p.W[t] + ch * n;
      const int off = d_COL_OFF[t];
      for (int j = 0; j < n; ++j) row[off + j] = Wt[j];
      row[KBIAS + t] = p.b[t][ch];     // bias as one-hot GEMM column
      // packed gather metadata: tbl(5) | f(4) | width(9) | chan_off(rest)
      int s = 0, start = 0;
      while (s < d_ND[t] - 1 && ch >= start + d_DW[t][s]) { start += d_DW[t][s]; ++s; }
      META[t * 256 + ch] = (unsigned)(d_TBASE[t] + s)
                         | ((unsigned)d_DF[t][s] << 5)
                         | ((unsigned)d_DW[t][s] << 9)
                         | ((unsigned)(ch - start) << 18);
    }
    if (tid < 21) EP[tid] = (unsigned long long)(size_t)p.emb[tid];
  }
  __syncthreads();

  const int gridx = gridDim.x;
  for (int tile = blockIdx.x; tile < p.ntiles; tile += gridx) {
    if (tile * 16 + 16 <= p.E)   // block-uniform: one branch per tile
      tile_body<true >(p, tile, gridx, WL, X, AI, TY, META, EP, tid, wave, half, l16);
    else
      tile_body<false>(p, tile, gridx, WL, X, AI, TY, META, EP, tid, wave, half, l16);
  }
}

extern "C" void kernel_launch(void* const* d_in, const int* in_sizes, int n_in,
                              void* d_out, int out_size, void* d_ws, size_t ws_size,
                              hipStream_t stream) {
  (void)d_ws; (void)ws_size; (void)n_in; (void)out_size;

  EncParams p;
  p.edge_attr = (const float*)d_in[0];
  p.out       = (float*)d_out;
  p.E         = in_sizes[0] / 10;

  // Flattening-order sniff: sorted dict keys => W first (in_sizes[1] == 256*4 = 1024);
  // insertion order => emb[t0][f0] first (15*52 = 780).
  const bool sorted_order = (in_sizes[1] == 1024);
  if (sorted_order) {
    for (int t = 0; t < 9;  ++t) p.W[t]   = (const float*)d_in[1 + t];
    for (int t = 0; t < 9;  ++t) p.b[t]   = (const float*)d_in[10 + t];
    for (int k = 0; k < 21; ++k) p.emb[k] = (const float*)d_in[19 + k];
  } else {
    for (int k = 0; k < 21; ++k) p.emb[k] = (const float*)d_in[1 + k];
    for (int t = 0; t < 9;  ++t) p.W[t]   = (const float*)d_in[22 + t];
    for (int t = 0; t < 9;  ++t) p.b[t]   = (const float*)d_in[31 + t];
  }

  p.ntiles = (p.E + 15) / 16;
  int grid = p.ntiles < 4096 ? p.ntiles : 4096;

  size_t shmem = (size_t)L_TOTAL_DW * 4;   // ~91 KB of 320 KB/WGP -> 3 blocks/WGP

  HomographEdgeEncoder_wmma_kernel<<<dim3(grid), dim3(256), shmem, stream>>>(p);
}